// BatchShapingLossModuleOld_68745246540073
// MI455X (gfx1250) — compile-verified
//
#include <hip/hip_runtime.h>
#include <cstdint>

#define NROWS 16384
#define NCOLS 2048
#define TPB   1024
#define EPT   (NROWS / TPB)   // 16 elements per thread

typedef float        v2f __attribute__((ext_vector_type(2)));
typedef float        v8f __attribute__((ext_vector_type(8)));
typedef unsigned int v4u __attribute__((ext_vector_type(4)));
typedef int          v4i __attribute__((ext_vector_type(4)));
typedef int          v8i __attribute__((ext_vector_type(8)));

__device__ __forceinline__ float rcp_f(float v) { return __builtin_amdgcn_rcpf(v); }

// Lentz continued fraction for the incomplete beta (fixed 24 CF steps).
__device__ float betacf(float a, float b, float x) {
    const float FPMIN = 1e-30f;
    float qab = a + b, qap = a + 1.0f, qam = a - 1.0f;
    float c = 1.0f;
    float d = 1.0f - qab * x * rcp_f(qap);
    if (fabsf(d) < FPMIN) d = FPMIN;
    d = rcp_f(d);
    float h = d;
    #pragma unroll
    for (int m = 1; m <= 12; ++m) {
        float fm = (float)m, m2 = 2.0f * fm;
        float aa = fm * (b - fm) * x * rcp_f((qam + m2) * (a + m2));
        d = 1.0f + aa * d;        if (fabsf(d) < FPMIN) d = FPMIN;
        c = 1.0f + aa * rcp_f(c); if (fabsf(c) < FPMIN) c = FPMIN;
        d = rcp_f(d);
        h *= d * c;
        aa = -(a + fm) * (qab + fm) * x * rcp_f((a + m2) * (qap + m2));
        d = 1.0f + aa * d;        if (fabsf(d) < FPMIN) d = FPMIN;
        c = 1.0f + aa * rcp_f(c); if (fabsf(c) < FPMIN) c = FPMIN;
        d = rcp_f(d);
        h *= d * c;
    }
    return h;
}

// I_x(0.6, 0.4); 1/B(0.6,0.4) = sin(0.4*pi)/pi
__device__ float betainc06_04(float x) {
    const float a = 0.6f, b = 0.4f;
    const float invBeta = 0.30272931f;           // 1 / 3.3032660
    float bt = __expf(a * __logf(x) + b * __logf(1.0f - x)) * invBeta;
    if (x < 0.8f)                                 // fixed split 0.8; (a+1)/(a+b+2) = 0.533
        return bt * betacf(a, b, x) * (1.0f / 0.6f);
    else
        return 1.0f - bt * betacf(b, a, 1.0f - x) * (1.0f / 0.4f);
}

__global__ void __launch_bounds__(TPB)
bsl_column_kernel(const float* __restrict__ x, float* __restrict__ part) {
    __shared__ float sm[NROWS];                   // 64 KB of the 320 KB WGP LDS
    const int tid = threadIdx.x;
    const int col = blockIdx.x;

    // ---- Stage the strided column via the Tensor Data Mover -------------
    // Tile: 1 x 16384 elements, dim0 stride = H (column-major gather),
    // contiguous into LDS. Issued once by wave 0 (EXEC ignored by TDM).
    if (tid == 0) {
        unsigned long long ga = (unsigned long long)(uintptr_t)(x) +
                                (unsigned long long)col * 4ull;
        uint32_t lds = (uint32_t)(uintptr_t)&sm[0];

        v4u g0;
        g0[0] = 1u;                               // count=1, user mode, no gather
        g0[1] = lds;                              // lds_addr (bytes)
        g0[2] = (uint32_t)ga;                     // global_addr[31:0]
        g0[3] = ((uint32_t)(ga >> 32) & 0x1FFFFFFu) | (2u << 30); // addr[56:32], type=2

        v8i g1;
        g1[0] = (int)(2u << 16);                  // data_size = 4 bytes; mask=0
        g1[1] = (int)(1u << 16);                  // tensor_dim0 = 1 (bits 63:48)
        g1[2] = (int)((NROWS & 0xFFFF) << 16);    // tensor_dim1 lo16 (bits 95:80)
        g1[3] = (int)(1u << 16);                  // tile_dim0 = 1 (bits 127:112)
        g1[4] = (int)NROWS;                       // tile_dim1 = 16384; tile_dim2 = 0
        g1[5] = (int)NCOLS;                       // tensor_dim0_stride = H
        g1[6] = 0;
        g1[7] = 0;

        v4i z4 = (v4i)0;
#if __clang_major__ >= 23
        v8i z8 = (v8i)0;
        __builtin_amdgcn_tensor_load_to_lds(g0, g1, z4, z4, z8, 0);
#else
        __builtin_amdgcn_tensor_load_to_lds(g0, g1, z4, z4, 0);
#endif
        __builtin_amdgcn_s_wait_tensorcnt(0);
    }
    __syncthreads();

    // ---- Elementwise betainc (monotone => order-preserving), in place ---
    #pragma unroll
    for (int s = 0; s < EPT; ++s) {
        int i = tid + s * TPB;
        sm[i] = betainc06_04(sm[i]);
    }
    __syncthreads();

    // ---- Bitonic sort ascending over 16384 elements in LDS --------------
    for (int k = 2; k <= NROWS; k <<= 1) {
        for (int j = k >> 1; j > 0; j >>= 1) {
            #pragma unroll
            for (int s = 0; s < EPT; ++s) {
                int i   = tid + s * TPB;
                int ixj = i ^ j;
                if (ixj > i) {
                    float va = sm[i], vb = sm[ixj];
                    bool  up = ((i & k) == 0);
                    if (up ? (va > vb) : (va < vb)) { sm[i] = vb; sm[ixj] = va; }
                }
            }
            __syncthreads();
        }
    }

    // ---- Squared error vs empirical CDF ---------------------------------
    const float inv = 1.0f / (float)(NROWS + 1);
    float acc = 0.0f;
    #pragma unroll
    for (int s = 0; s < EPT; ++s) {
        int i = tid + s * TPB;
        float d = sm[i] - (float)(i + 1) * inv;
        acc += d * d;
    }
    __syncthreads();   // done reading sorted data; sm reused for reduction

    // wave32 reduce: 1024 partials -> 32 wave sums
    #pragma unroll
    for (int off = 16; off > 0; off >>= 1) acc += __shfl_down(acc, off);
    if ((tid & 31) == 0) sm[tid >> 5] = acc;      // sm[0..31]
    if (tid >= 32 && tid < 64) sm[tid] = 0.0f;    // zero-pad to 64 for 16x4 A
    __syncthreads();

    // ---- Exact f32 block sum via one WMMA (A 16x4 f32, B = ones) --------
    if (tid < 32) {                               // wave 0, EXEC all ones
        v2f A;  A.x = sm[2 * tid]; A.y = sm[2 * tid + 1];
        v2f Bv; Bv.x = 1.0f; Bv.y = 1.0f;
        v8f C = {0.0f, 0.0f, 0.0f, 0.0f, 0.0f, 0.0f, 0.0f, 0.0f};
        C = __builtin_amdgcn_wmma_f32_16x16x4_f32(
                /*neg_a=*/false, A, /*neg_b=*/false, Bv,
                /*c_mod=*/(short)0, C, /*reuse_a=*/false, /*reuse_b=*/false);
        float sgm = C[0] + C[1] + C[2] + C[3] + C[4] + C[5] + C[6] + C[7];
        float t0  = __shfl(sgm, 0);               // rows M=0..7  at N=0
        float t16 = __shfl(sgm, 16);              // rows M=8..15 at N=0
        if (tid == 0) part[blockIdx.x] = t0 + t16;
    }
}

__global__ void __launch_bounds__(TPB)
bsl_reduce_kernel(const float* __restrict__ part, float* __restrict__ out) {
    __shared__ float red[32];
    int t = threadIdx.x;
    float v = part[t] + part[t + TPB];            // 2048 partials, 2 per thread
    #pragma unroll
    for (int off = 16; off > 0; off >>= 1) v += __shfl_down(v, off);
    if ((t & 31) == 0) red[t >> 5] = v;
    __syncthreads();
    if (t < 32) {
        float w = red[t];
        #pragma unroll
        for (int off = 16; off > 0; off >>= 1) w += __shfl_down(w, off);
        if (t == 0) out[0] = w * (1.0f / (float)NROWS);   // loss = sum / n
    }
}

extern "C" void kernel_launch(void* const* d_in, const int* in_sizes, int n_in,
                              void* d_out, int out_size, void* d_ws, size_t ws_size,
                              hipStream_t stream) {
    const float* x    = (const float*)d_in[0];
    float*       part = (float*)d_ws;             // 2048 floats, fully rewritten each call
    float*       out  = (float*)d_out;
    bsl_column_kernel<<<NCOLS, TPB, 0, stream>>>(x, part);
    bsl_reduce_kernel<<<1, TPB, 0, stream>>>(part, out);
}